// TemporalGNN_21114059227634
// MI455X (gfx1250) — compile-verified
//
#include <hip/hip_runtime.h>
#include <hip/hip_bf16.h>

// ---------------------------------------------------------------------------
// GCN (2 layers + linear head) for MI455X / gfx1250.
// F = H = 64 fixed. N divisible by 16 (N=100000 -> 6250 WMMA row-tiles).
// GEMMs use V_WMMA_F32_16X16X4_F32 (exact fp32). Aggregation is L2-resident
// gather + global_atomic_add_f32 scatter (192MB L2 holds the whole problem).
// ---------------------------------------------------------------------------

typedef float v2f __attribute__((ext_vector_type(2)));
typedef float v8f __attribute__((ext_vector_type(8)));

#define FEAT 64

// ---- degree / norm ---------------------------------------------------------

__global__ void k_deg_init(float* deg, int n) {
    int i = blockIdx.x * blockDim.x + threadIdx.x;
    if (i < n) deg[i] = 1.0f;   // self-loop contributes 1
}

__global__ void k_deg_count(const int* __restrict__ dst, float* deg, int e) {
    int i = blockIdx.x * blockDim.x + threadIdx.x;
    if (i < e) atomicAdd(&deg[dst[i]], 1.0f);
}

__global__ void k_dinv(float* deg, int n) {
    int i = blockIdx.x * blockDim.x + threadIdx.x;
    if (i < n) deg[i] = rsqrtf(deg[i]);   // deg >= 1 always (self loops)
}

// ---- GEMM: H[N,64] = X[N,64] @ W[64,64] via V_WMMA_F32_16X16X4_F32 ---------
// One wave computes a 16-row x 64-col stripe: 4 accumulators, 16 K-steps.
// A layout (16x4 f32): lane<16 -> K={0,1}, lane>=16 -> K={2,3}, M=lane&15.
// B layout (4x16 f32): mirrored K split, N=lane&15.
// C/D (16x16 f32, 8 VGPRs): lane<16 -> M=j, lane>=16 -> M=j+8, N=lane&15.

__global__ void k_gemm64(const float* __restrict__ X, const float* __restrict__ W,
                         float* __restrict__ Hout, int nTiles) {
    __shared__ float sW[FEAT * FEAT];
    for (int i = threadIdx.x; i < FEAT * FEAT; i += blockDim.x) sW[i] = W[i];
    __syncthreads();

    const int wavesPerBlock = blockDim.x >> 5;
    const int wave = blockIdx.x * wavesPerBlock + (threadIdx.x >> 5);
    if (wave >= nTiles) return;              // wave-uniform: EXEC stays all-1s

    const int lane = threadIdx.x & 31;
    const int mrow = wave * 16 + (lane & 15);
    const int kb   = (lane >> 4) * 2;        // K sub-offset within 4-wide step
    const int ncol = lane & 15;

    v8f acc0 = {}, acc1 = {}, acc2 = {}, acc3 = {};

#pragma unroll
    for (int kk = 0; kk < 16; ++kk) {
        const int k = kk * 4 + kb;
        v2f a;
        a.x = X[mrow * FEAT + k];
        a.y = X[mrow * FEAT + k + 1];

        v2f b0, b1, b2, b3;
        b0.x = sW[(k) * FEAT + ncol +  0]; b0.y = sW[(k + 1) * FEAT + ncol +  0];
        b1.x = sW[(k) * FEAT + ncol + 16]; b1.y = sW[(k + 1) * FEAT + ncol + 16];
        b2.x = sW[(k) * FEAT + ncol + 32]; b2.y = sW[(k + 1) * FEAT + ncol + 32];
        b3.x = sW[(k) * FEAT + ncol + 48]; b3.y = sW[(k + 1) * FEAT + ncol + 48];

        acc0 = __builtin_amdgcn_wmma_f32_16x16x4_f32(false, a, false, b0, (short)0, acc0, false, false);
        acc1 = __builtin_amdgcn_wmma_f32_16x16x4_f32(false, a, false, b1, (short)0, acc1, false, false);
        acc2 = __builtin_amdgcn_wmma_f32_16x16x4_f32(false, a, false, b2, (short)0, acc2, false, false);
        acc3 = __builtin_amdgcn_wmma_f32_16x16x4_f32(false, a, false, b3, (short)0, acc3, false, false);
    }

    const int rbase = wave * 16 + (lane >> 4) * 8;
#pragma unroll
    for (int j = 0; j < 8; ++j) {
        Hout[(rbase + j) * FEAT + ncol +  0] = acc0[j];
        Hout[(rbase + j) * FEAT + ncol + 16] = acc1[j];
        Hout[(rbase + j) * FEAT + ncol + 32] = acc2[j];
        Hout[(rbase + j) * FEAT + ncol + 48] = acc3[j];
    }
}

// ---- self-loop term: out[i] = dinv[i]^2 * h[i] -----------------------------

__global__ void k_selfinit(const float* __restrict__ Hin, const float* __restrict__ dinv,
                           float* __restrict__ Out, int n) {
    int t = blockIdx.x * blockDim.x + threadIdx.x;
    if (t >= n * FEAT) return;
    int node = t >> 6;
    float dv = dinv[node];
    Out[t] = Hin[t] * (dv * dv);
}

// ---- edge scatter: out[dst] += dinv[src]*dinv[dst] * h[src] ----------------
// One wave per edge; lane handles features {2*lane, 2*lane+1} (float2 load,
// two global_atomic_add_f32). Data is L2-resident (working set < 192MB).

__global__ void k_scatter(const float* __restrict__ Hin, const int* __restrict__ src,
                          const int* __restrict__ dst, const float* __restrict__ dinv,
                          float* __restrict__ Out, int e) {
    long long gid = (long long)blockIdx.x * blockDim.x + threadIdx.x;
    int edge = (int)(gid >> 5);
    if (edge >= e) return;
    int lane = (int)(gid & 31);

    int s = src[edge];
    int d = dst[edge];
    float nrm = dinv[s] * dinv[d];

    const float2* hs = (const float2*)(Hin + (size_t)s * FEAT);
    float2 v = hs[lane];
    atomicAdd(&Out[(size_t)d * FEAT + lane * 2 + 0], v.x * nrm);
    atomicAdd(&Out[(size_t)d * FEAT + lane * 2 + 1], v.y * nrm);
}

// ---- bias + relu -----------------------------------------------------------

__global__ void k_bias_relu(float* __restrict__ Out, const float* __restrict__ b, int n) {
    int t = blockIdx.x * blockDim.x + threadIdx.x;
    if (t >= n * FEAT) return;
    float v = Out[t] + b[t & (FEAT - 1)];
    Out[t] = v > 0.0f ? v : 0.0f;
}

// ---- head: out[i] = dot(h2[i], Wl) + bl ------------------------------------
// One wave32 per node: each lane covers 2 features, shfl-xor reduce.

__global__ void k_head(const float* __restrict__ Hin, const float* __restrict__ Wl,
                       const float* __restrict__ bl, float* __restrict__ Out, int n) {
    long long gid = (long long)blockIdx.x * blockDim.x + threadIdx.x;
    int node = (int)(gid >> 5);
    if (node >= n) return;
    int lane = (int)(gid & 31);

    const float* hrow = Hin + (size_t)node * FEAT;
    float sum = hrow[lane] * Wl[lane] + hrow[lane + 32] * Wl[lane + 32];
#pragma unroll
    for (int off = 16; off > 0; off >>= 1)
        sum += __shfl_xor(sum, off, 32);
    if (lane == 0) Out[node] = sum + bl[0];
}

// ---------------------------------------------------------------------------

extern "C" void kernel_launch(void* const* d_in, const int* in_sizes, int n_in,
                              void* d_out, int out_size, void* d_ws, size_t ws_size,
                              hipStream_t stream) {
    const float* x   = (const float*)d_in[0];
    const int*   ei  = (const int*)d_in[1];
    const float* W1  = (const float*)d_in[2];
    const float* b1  = (const float*)d_in[3];
    const float* W2  = (const float*)d_in[4];
    const float* b2  = (const float*)d_in[5];
    const float* Wl  = (const float*)d_in[6];
    const float* bl  = (const float*)d_in[7];

    const int N = in_sizes[0] / FEAT;      // 100000
    const int E = in_sizes[1] / 2;         // 3200000
    const int* src = ei;
    const int* dst = ei + E;

    // workspace layout (floats): [dinv: N][bufA: N*64][bufB: N*64]
    float* dinv = (float*)d_ws;
    float* bufA = dinv + N;
    float* bufB = bufA + (size_t)N * FEAT;

    const int T = 256;
    const int gN   = (N + T - 1) / T;
    const int gE   = (E + T - 1) / T;
    const int gNF  = (N * FEAT + T - 1) / T;
    const int gE32 = (int)(((long long)E * 32 + T - 1) / T);
    const int gN32 = (int)(((long long)N * 32 + T - 1) / T);
    const int nTiles = N / 16;                       // 6250
    const int gemmBlocks = (nTiles + (T / 32) - 1) / (T / 32);

    // normalization: deg (with self loop) -> rsqrt, in place
    k_deg_init <<<gN, T, 0, stream>>>(dinv, N);
    k_deg_count<<<gE, T, 0, stream>>>(dst, dinv, E);
    k_dinv     <<<gN, T, 0, stream>>>(dinv, N);

    // ---- layer 1 ----
    k_gemm64   <<<gemmBlocks, T, 0, stream>>>(x, W1, bufA, nTiles);
    k_selfinit <<<gNF, T, 0, stream>>>(bufA, dinv, bufB, N);
    k_scatter  <<<gE32, T, 0, stream>>>(bufA, src, dst, dinv, bufB, E);
    k_bias_relu<<<gNF, T, 0, stream>>>(bufB, b1, N);

    // ---- layer 2 ----
    k_gemm64   <<<gemmBlocks, T, 0, stream>>>(bufB, W2, bufA, nTiles);
    k_selfinit <<<gNF, T, 0, stream>>>(bufA, dinv, bufB, N);
    k_scatter  <<<gE32, T, 0, stream>>>(bufA, src, dst, dinv, bufB, E);
    k_bias_relu<<<gNF, T, 0, stream>>>(bufB, b2, N);

    // ---- head ----
    k_head     <<<gN32, T, 0, stream>>>(bufB, Wl, bl, (float*)d_out, N);
}